// LightGCN_43370579755264
// MI455X (gfx1250) — compile-verified
//
#include <hip/hip_runtime.h>
#include <hip/hip_bf16.h>
#include <stdint.h>

#define EMB    64
#define BLOCK  256
#define CHUNK  1024
#define WAVES  (BLOCK / 32)
#define TILES_PER_WAVE (CHUNK / 16 / WAVES)   // 8 tiles of 16 nnz per wave

typedef __attribute__((ext_vector_type(2))) float v2f;
typedef __attribute__((ext_vector_type(8))) float v8f;

// ---------------------------------------------------------------------------
// E0 = concat(user_w, item_w); Ecur = E0; Eacc = E0
// ---------------------------------------------------------------------------
__global__ void init_concat(const float4* __restrict__ u, const float4* __restrict__ it,
                            float4* __restrict__ Ecur, float4* __restrict__ Eacc,
                            int nU4, int nTot4) {
    int i = blockIdx.x * blockDim.x + threadIdx.x;
    if (i >= nTot4) return;
    float4 v = (i < nU4) ? u[i] : it[i - nU4];
    Ecur[i] = v;
    Eacc[i] = v;
}

__global__ void zero_f4(float4* __restrict__ p, int n4) {
    int i = blockIdx.x * blockDim.x + threadIdx.x;
    if (i < n4) p[i] = make_float4(0.f, 0.f, 0.f, 0.f);
}

__global__ void acc_add(float4* __restrict__ acc, const float4* __restrict__ x, int n4) {
    int i = blockIdx.x * blockDim.x + threadIdx.x;
    if (i < n4) {
        float4 a = acc[i], b = x[i];
        a.x += b.x; a.y += b.y; a.z += b.z; a.w += b.w;
        acc[i] = a;
    }
}

// ---------------------------------------------------------------------------
// COO SpMM: Enxt[r,:] += val * Ecur[c,:]
// Per 16-nnz tile: msgs = diag(val_16) x Erows(16x64), done as 4 chained
// V_WMMA_F32_16X16X4_F32 per 16-column tile, then per-lane atomic scatter.
// COO chunks are staged global->LDS with async b128 loads (ASYNCcnt).
// ---------------------------------------------------------------------------
__global__ __launch_bounds__(BLOCK)
void spmm_coo_wmma(const int* __restrict__ rows, const int* __restrict__ cols,
                   const float* __restrict__ vals, const float* __restrict__ Ecur,
                   float* __restrict__ Enxt, int nnz, int nChunks) {
    __shared__ int   s_row[CHUNK];
    __shared__ int   s_col[CHUNK];
    __shared__ float s_val[CHUNK];

    const int tid  = threadIdx.x;
    const int wave = tid >> 5;
    const int lane = tid & 31;
    const int half = lane >> 4;   // 0: lanes 0-15, 1: lanes 16-31
    const int m    = lane & 15;

    for (int ch = blockIdx.x; ch < nChunks; ch += gridDim.x) {
        const int base = ch * CHUNK;

        if (base + CHUNK <= nnz) {
            // ---- async stage: 3 x b128 per thread (12 KB total) ----
            const unsigned voff = (unsigned)(base + tid * 4) * 4u;     // byte offset
            const unsigned ldsR = (unsigned)(size_t)s_row + (unsigned)tid * 16u;
            const unsigned ldsC = (unsigned)(size_t)s_col + (unsigned)tid * 16u;
            const unsigned ldsV = (unsigned)(size_t)s_val + (unsigned)tid * 16u;
            asm volatile("global_load_async_to_lds_b128 %0, %1, %2"
                         :: "v"(ldsR), "v"(voff),
                            "s"((unsigned long long)(size_t)rows) : "memory");
            asm volatile("global_load_async_to_lds_b128 %0, %1, %2"
                         :: "v"(ldsC), "v"(voff),
                            "s"((unsigned long long)(size_t)cols) : "memory");
            asm volatile("global_load_async_to_lds_b128 %0, %1, %2"
                         :: "v"(ldsV), "v"(voff),
                            "s"((unsigned long long)(size_t)vals) : "memory");
            // prefetch this block's next chunk while we wait
            const int nb = base + (int)gridDim.x * CHUNK;
            if (nb + CHUNK <= nnz) {
                __builtin_prefetch(rows + nb, 0, 0);
                __builtin_prefetch(cols + nb, 0, 0);
                __builtin_prefetch(vals + nb, 0, 0);
            }
            asm volatile("s_wait_asynccnt 0x0" ::: "memory");
        } else {
            // ---- tail chunk: guarded scalar stage, pad with zeros ----
            for (int t = tid; t < CHUNK; t += BLOCK) {
                int g = base + t;
                bool ok = g < nnz;
                s_row[t] = ok ? rows[g] : 0;
                s_col[t] = ok ? cols[g] : 0;
                s_val[t] = ok ? vals[g] : 0.0f;
            }
        }
        __syncthreads();

        #pragma unroll
        for (int t = 0; t < TILES_PER_WAVE; ++t) {
            const int o = (wave * TILES_PER_WAVE + t) * 16;
            const float vm = s_val[o + m];
            v8f D0 = {0.f,0.f,0.f,0.f,0.f,0.f,0.f,0.f};
            v8f D1 = D0, D2 = D0, D3 = D0;

            #pragma unroll
            for (int kk = 0; kk < 4; ++kk) {
                // A (16x4 f32): VGPR0 holds K=half*2, VGPR1 holds K=half*2+1, M=m.
                // diag(val): A[M][K] = v_M iff global k (kk*4+K) == M.
                const int j0 = kk * 4 + half * 2;
                v2f A;
                A.x = (m == j0    ) ? vm : 0.0f;
                A.y = (m == j0 + 1) ? vm : 0.0f;
                // B (4x16 f32): VGPR0 row K=half*2, VGPR1 row K=half*2+1, N=m.
                const int c0 = s_col[o + j0]     * EMB;
                const int c1 = s_col[o + j0 + 1] * EMB;
                v2f B0 = { Ecur[c0 +  0 + m], Ecur[c1 +  0 + m] };
                v2f B1 = { Ecur[c0 + 16 + m], Ecur[c1 + 16 + m] };
                v2f B2 = { Ecur[c0 + 32 + m], Ecur[c1 + 32 + m] };
                v2f B3 = { Ecur[c0 + 48 + m], Ecur[c1 + 48 + m] };
                D0 = __builtin_amdgcn_wmma_f32_16x16x4_f32(false, A, false, B0, (short)0, D0, false, false);
                D1 = __builtin_amdgcn_wmma_f32_16x16x4_f32(false, A, false, B1, (short)0, D1, false, false);
                D2 = __builtin_amdgcn_wmma_f32_16x16x4_f32(false, A, false, B2, (short)0, D2, false, false);
                D3 = __builtin_amdgcn_wmma_f32_16x16x4_f32(false, A, false, B3, (short)0, D3, false, false);
            }

            // D layout: VGPR g -> row M = g + half*8, col N = m (per 16-col tile)
            #pragma unroll
            for (int g = 0; g < 8; ++g) {
                const int r = s_row[o + half * 8 + g] * EMB;
                unsafeAtomicAdd(&Enxt[r +  0 + m], D0[g]);
                unsafeAtomicAdd(&Enxt[r + 16 + m], D1[g]);
                unsafeAtomicAdd(&Enxt[r + 32 + m], D2[g]);
                unsafeAtomicAdd(&Enxt[r + 48 + m], D3[g]);
            }
        }
        __syncthreads();
    }
}

// ---------------------------------------------------------------------------
// out = 0.25 * concat(Eacc[users], Eacc[N_USER+pos], Eacc[N_USER+neg])
// ---------------------------------------------------------------------------
__global__ void gather_out(const float2* __restrict__ Eacc2, const int* __restrict__ users,
                           const int* __restrict__ pos, const int* __restrict__ neg,
                           float2* __restrict__ out2, int batch, int nUser) {
    int i = blockIdx.x * blockDim.x + threadIdx.x;
    int per = batch * (EMB / 2);
    if (i >= 3 * per) return;
    int sec = i / per;
    int rem = i - sec * per;
    int b = rem >> 5;       // EMB/2 == 32
    int e = rem & 31;
    int node = (sec == 0) ? users[b]
             : (sec == 1) ? nUser + pos[b]
                          : nUser + neg[b];
    float2 v = Eacc2[node * (EMB / 2) + e];
    v.x *= 0.25f; v.y *= 0.25f;
    out2[i] = v;
}

// ---------------------------------------------------------------------------
extern "C" void kernel_launch(void* const* d_in, const int* in_sizes, int n_in,
                              void* d_out, int out_size, void* d_ws, size_t ws_size,
                              hipStream_t stream) {
    const float* user_w = (const float*)d_in[0];
    const float* item_w = (const float*)d_in[1];
    const int*   rows   = (const int*)  d_in[2];
    const int*   cols   = (const int*)  d_in[3];
    const float* vals   = (const float*)d_in[4];
    const int*   users  = (const int*)  d_in[5];
    const int*   pos    = (const int*)  d_in[6];
    const int*   neg    = (const int*)  d_in[7];

    const int nU    = in_sizes[0] / EMB;
    const int nI    = in_sizes[1] / EMB;
    const int nTot  = nU + nI;
    const int nnz   = in_sizes[2];
    const int batch = in_sizes[5];

    float* Ecur = (float*)d_ws;
    float* Enxt = Ecur + (size_t)nTot * EMB;
    float* Eacc = Enxt + (size_t)nTot * EMB;

    const int n4 = nTot * (EMB / 4);
    init_concat<<<(n4 + 255) / 256, 256, 0, stream>>>(
        (const float4*)user_w, (const float4*)item_w,
        (float4*)Ecur, (float4*)Eacc, nU * (EMB / 4), n4);

    const int nChunks = (nnz + CHUNK - 1) / CHUNK;
    const int grid = nChunks < 2048 ? nChunks : 2048;

    for (int l = 0; l < 3; ++l) {
        zero_f4<<<(n4 + 255) / 256, 256, 0, stream>>>((float4*)Enxt, n4);
        spmm_coo_wmma<<<grid, BLOCK, 0, stream>>>(rows, cols, vals, Ecur, Enxt, nnz, nChunks);
        acc_add<<<(n4 + 255) / 256, 256, 0, stream>>>((float4*)Eacc, (const float4*)Enxt, n4);
        float* tmp = Ecur; Ecur = Enxt; Enxt = tmp;
    }

    const int total2 = 3 * batch * (EMB / 2);
    gather_out<<<(total2 + 255) / 256, 256, 0, stream>>>(
        (const float2*)Eacc, users, pos, neg, (float2*)d_out, batch, nU);
}